// PlanFormer_18992345383477
// MI455X (gfx1250) — compile-verified
//
#include <hip/hip_runtime.h>
#include <hip/hip_bf16.h>

// ---------------------------------------------------------------------------
// PlanFormer for MI455X (gfx1250, wave32).
//
// Key algebraic reductions (exact, not approximations):
//  * mha with Q=1: softmax over a length-1 axis == 1.0 exactly -> out = (q@Wv+bv)@Wo+bo.
//  * deform_attn: projection/bilinear/attn-sum are linear, so we gather the
//    <=16 needed (bev+PE) rows per (batch,head), combine with scalar weights
//    attn*w_corner*valid, then ONE 256x32 matvec per head + dbv * sum(weights).
//    This removes the 2.5e11-FLOP dense value projection entirely.
//
// All projections run as V_WMMA_F32_16X16X4_F32 with M = 16 batches in one
// workgroup (8 waves). FF=512 GEMMs are split into two K=256 halves with
// C-accumulation so LDS stays at 4 x 16KB static buffers.
// ---------------------------------------------------------------------------

typedef float v2f __attribute__((ext_vector_type(2)));
typedef float v8f __attribute__((ext_vector_type(8)));

#define NBATCH 16
#define DMODEL 256
#define NHEADS 8
#define HDIM   32
#define NPTS   4
#define NFF    512
#define NLAYER 3
#define NMODES 6
#define NT     6
#define GH     200
#define GW     200

struct LayerP {
  const float *Wv, *bv, *Wo, *bo, *g1, *be1;
  const float *dWv, *dbv, *dWo, *dbo, *dWa, *dba, *dWu, *dbu, *g2, *be2;
  const float *mW1, *mb1, *mW2, *mb2;
};

struct Params {
  const int*   cmd;      // (16,1) int32
  const float* bev;      // (16, 40000, 256)
  const float* track;    // (16, 256)
  const float* motion;   // (16, 6, 256)
  const float* cmd_emb;  // (6, 256)
  const float* plan_pos; // (1,1,256)
  const float *pq_W1, *pq_b1, *pq_W2, *pq_b2;
  const float *tr_W1, *tr_b1, *tr_W2, *tr_b2;
  LayerP lyr[NLAYER];
  float* out_q;          // (16, 256)
  float* out_traj;       // (16, 6, 2)
};

enum { GF_RELU = 1, GF_ACC = 2 };

// C(16 x 16*ntiles) = A(16x256, LDS row-major) @ W(256 x ., global, ldw) [+bias | +=C]
// using V_WMMA_F32_16X16X4_F32. One N-tile per wave iteration; wave-uniform
// control flow keeps EXEC all-ones at every WMMA as the ISA requires.
__device__ inline void gemm16(const float* __restrict__ A,
                              const float* __restrict__ W, int ldw, int ntiles,
                              const float* __restrict__ bias,
                              float* __restrict__ C, int ldc, int flags) {
  const int lane = threadIdx.x & 31;
  const int wave = threadIdx.x >> 5;
  const int col  = lane & 15;   // N index inside tile; also M row for A frag
  const int hi   = lane >> 4;   // 0: K=0,1  1: K=2,3 (ISA 16x4 f32 A/B layout)
  for (int nt = wave; nt < ntiles; nt += 8) {
    const int n0 = nt << 4;
    v8f acc;
    if (flags & GF_ACC) {
#pragma unroll
      for (int r = 0; r < 8; ++r) acc[r] = C[(r + hi * 8) * ldc + n0 + col];
    } else {
      const float bv = bias ? bias[n0 + col] : 0.0f;
#pragma unroll
      for (int r = 0; r < 8; ++r) acc[r] = bv;
    }
    const float* Ap = A + col * 256 + hi * 2;           // A[M=col][K=k0+hi*2+{0,1}]
    const float* Wp = W + (hi * 2) * ldw + n0 + col;    // B[K=k0+hi*2+{0,1}][N=col]
#pragma unroll 4
    for (int k0 = 0; k0 < 256; k0 += 4) {
      v2f av, wv;
      av.x = Ap[k0 + 0];
      av.y = Ap[k0 + 1];
      wv.x = Wp[k0 * ldw];
      wv.y = Wp[(k0 + 1) * ldw];
      acc = __builtin_amdgcn_wmma_f32_16x16x4_f32(false, av, false, wv,
                                                  (short)0, acc, false, false);
    }
#pragma unroll
    for (int r = 0; r < 8; ++r) {
      float v = acc[r];
      if (flags & GF_RELU) v = fmaxf(v, 0.0f);
      C[(r + hi * 8) * ldc + n0 + col] = v;   // C[M=r+hi*8][N=col] per ISA C/D layout
    }
  }
}

// out = LN(X + R) * g + be   (row-wise over 256), out may alias X.
__device__ inline void layernorm_res(const float* X, const float* R,
                                     const float* g, const float* be,
                                     float* OUT, float* MEAN, float* RSTD) {
  const int tid = threadIdx.x;
  if (tid < NBATCH) {
    float s = 0.f, s2 = 0.f;
    for (int c = 0; c < DMODEL; ++c) {
      float v = X[tid * DMODEL + c] + R[tid * DMODEL + c];
      s += v; s2 += v * v;
    }
    float mu  = s * (1.0f / DMODEL);
    float var = s2 * (1.0f / DMODEL) - mu * mu;
    MEAN[tid] = mu;
    RSTD[tid] = rsqrtf(var + 1e-5f);
  }
  __syncthreads();
  for (int e = tid; e < NBATCH * DMODEL; e += 256) {
    int b = e >> 8, c = e & 255;
    float v = X[e] + R[e];
    OUT[e] = (v - MEAN[b]) * RSTD[b] * g[c] + be[c];
  }
  __syncthreads();
}

__global__ __launch_bounds__(256) void PlanFormer_18992345383477_kernel(Params P) {
  __shared__ float S_q[4096];   // persistent q / q1
  __shared__ float S_a[4096];   // staging A / gather buffer / FF half
  __shared__ float S_b[4096];   // staging B / sampled / q2
  __shared__ float S_u[4096];   // stage0 y-accumulator  |  layer-loop scratch

  // Scratch layout inside S_u during the layer loop (stage0 clobbers all of it):
  float* OFFb  = S_u;                     // 16x64  (written by gemm, read->regs)
  float* ALOGb = S_u + 1024;              // 16x32
  float* FXb   = S_u;                     // 512  (b,h,p) bilinear wx   (reuses OFF region)
  float* FYb   = S_u + 512;               // 512
  float* FAb   = S_u + 1024;              // 512  attn weight           (reuses ALOG region)
  int*   IXb   = (int*)(S_u + 1536);      // 512
  int*   IYb   = (int*)(S_u + 2048);      // 512
  float* WSUM  = S_u + 2560;              // 128  (b,h)
  float* SINV  = S_u + 2688;              // 64   10000^(-j/64)
  float* MEANb = S_u + 2752;              // 16
  float* RSTDb = S_u + 2768;              // 16
  float* TRAJb = S_u + 2784;              // 192

  const int tid = threadIdx.x;

  // ---------------- stage 0: plan-query MLP + max over modes ----------------
  for (int m = 0; m < NMODES; ++m) {
    for (int e = tid; e < NBATCH * DMODEL; e += 256) {
      int b = e >> 8, c = e & 255;
      int cm = P.cmd[b];
      S_a[e] = P.cmd_emb[cm * DMODEL + c] + P.track[e]
             + P.motion[(b * NMODES + m) * DMODEL + c];
    }
    __syncthreads();
    for (int half = 0; half < 2; ++half) {
      gemm16(S_a, P.pq_W1 + half * 256, NFF, 16, P.pq_b1 + half * 256,
             S_b, 256, GF_RELU);
      __syncthreads();
      gemm16(S_b, P.pq_W2 + half * 256 * 256, 256, 16,
             half == 0 ? P.pq_b2 : nullptr, S_u, 256, half == 0 ? 0 : GF_ACC);
      __syncthreads();
    }
    for (int e = tid; e < NBATCH * DMODEL; e += 256)
      S_q[e] = (m == 0) ? S_u[e] : fmaxf(S_q[e], S_u[e]);
    __syncthreads();
  }
  for (int e = tid; e < NBATCH * DMODEL; e += 256)
    S_q[e] += P.plan_pos[e & 255];
  // inv-frequency table for the 2D sincos PE: 10000^(-j/64) = 2^(-j*log2(1e4)/64)
  if (tid < 64)
    SINV[tid] = exp2f(-(float)tid * (13.287712379549449f / 64.0f));
  __syncthreads();

  // ---------------- transformer layers ----------------
  for (int l = 0; l < NLAYER; ++l) {
    const LayerP& Lp = P.lyr[l];

    // mha (Q=1 => softmax==1 => out = (q@Wv+bv)@Wo+bo exactly)
    gemm16(S_q, Lp.Wv, 256, 16, Lp.bv, S_a, 256, 0);
    __syncthreads();
    gemm16(S_a, Lp.Wo, 256, 16, Lp.bo, S_b, 256, 0);
    __syncthreads();
    layernorm_res(S_q, S_b, Lp.g1, Lp.be1, S_q, MEANb, RSTDb);  // q1 in S_q

    // deform-attn heads: offsets + attention logits
    gemm16(S_q, Lp.dWo, NHEADS * NPTS * 2, 4, Lp.dbo, OFFb, 64, 0);
    gemm16(S_q, Lp.dWa, NHEADS * NPTS, 2, Lp.dba, ALOGb, 32, 0);
    __syncthreads();

    // per-(b,h): softmax over P, bilinear coefficients. Copy own off/alog to
    // registers first because FX/FA alias the OFF/ALOG regions.
    float offr[8], alogr[4];
    if (tid < NBATCH * NHEADS) {
      int b = tid >> 3, h = tid & 7;
#pragma unroll
      for (int i = 0; i < 8; ++i) offr[i] = OFFb[b * 64 + h * 8 + i];
#pragma unroll
      for (int p = 0; p < 4; ++p) alogr[p] = ALOGb[b * 32 + h * 4 + p];
    }
    __syncthreads();
    if (tid < NBATCH * NHEADS) {
      float mx = fmaxf(fmaxf(alogr[0], alogr[1]), fmaxf(alogr[2], alogr[3]));
      float ex[4]; float sum = 0.f;
#pragma unroll
      for (int p = 0; p < 4; ++p) { ex[p] = expf(alogr[p] - mx); sum += ex[p]; }
      float inv_s = 1.0f / sum;
      float ws = 0.f;
#pragma unroll
      for (int p = 0; p < 4; ++p) {
        float at = ex[p] * inv_s;
        float lx = 0.5f + offr[p * 2 + 0] * (1.0f / GW);
        float ly = 0.5f + offr[p * 2 + 1] * (1.0f / GH);
        float x = lx * (float)GW - 0.5f;
        float y = ly * (float)GH - 0.5f;
        float x0f = floorf(x), y0f = floorf(y);
        float wx = x - x0f, wy = y - y0f;
        int xi = (int)x0f, yi = (int)y0f;
        int idx = tid * 4 + p;
        FXb[idx] = wx; FYb[idx] = wy; FAb[idx] = at; IXb[idx] = xi; IYb[idx] = yi;
        float sv = 0.f;
#pragma unroll
        for (int cnr = 0; cnr < 4; ++cnr) {
          int dx = cnr & 1, dy = cnr >> 1;
          int xx = xi + dx, yy = yi + dy;
          if (xx >= 0 && xx < GW && yy >= 0 && yy < GH)
            sv += (dx ? wx : 1.f - wx) * (dy ? wy : 1.f - wy);
        }
        ws += at * sv;
      }
      WSUM[tid] = ws;
    }
    __syncthreads();

    // per head: gather weighted (bev + PE) rows, then 256x32 matvec via WMMA
    for (int h = 0; h < NHEADS; ++h) {
      for (int e = tid; e < NBATCH * DMODEL; e += 256) {
        int b = e >> 8, c = e & 255;
        float invc = SINV[c & 63];
        bool useY = (c < 128);
        bool useSin = ((c & 127) < 64);
        float acc = 0.f;
        int base = (b * NHEADS + h) * 4;
#pragma unroll
        for (int p = 0; p < 4; ++p) {
          int idx = base + p;
          int xi = IXb[idx], yi = IYb[idx];
          float wx = FXb[idx], wy = FYb[idx], at = FAb[idx];
#pragma unroll
          for (int cnr = 0; cnr < 4; ++cnr) {
            int dx = cnr & 1, dy = cnr >> 1;
            int xx = xi + dx, yy = yi + dy;
            if (xx >= 0 && xx < GW && yy >= 0 && yy < GH) {
              float w = (dx ? wx : 1.f - wx) * (dy ? wy : 1.f - wy) * at;
              float bevv = P.bev[(long)(b * (GH * GW) + yy * GW + xx) * DMODEL + c];
              float pos = useY ? (float)yy : (float)xx;
              float ang = pos * invc;
              float pe = useSin ? sinf(ang) : cosf(ang);
              acc += w * (bevv + pe);
            }
          }
        }
        S_a[e] = acc;
      }
      __syncthreads();
      gemm16(S_a, Lp.dWv + h * HDIM, 256, 2, nullptr, S_b + h * HDIM, 256, 0);
      __syncthreads();
    }
    // + dbv * sum(attn*bilinear weights)   (bias through the valid mask)
    for (int e = tid; e < NBATCH * DMODEL; e += 256) {
      int b = e >> 8, c = e & 255;
      S_b[e] += Lp.dbv[c] * WSUM[b * NHEADS + (c >> 5)];
    }
    __syncthreads();
    gemm16(S_b, Lp.dWu, 256, 16, Lp.dbu, S_a, 256, 0);
    __syncthreads();
    layernorm_res(S_q, S_a, Lp.g2, Lp.be2, S_b, MEANb, RSTDb);  // q2 in S_b

    // FFN: FF=512 split into two K=256 halves with C-accumulation into S_q
    for (int half = 0; half < 2; ++half) {
      gemm16(S_b, Lp.mW1 + half * 256, NFF, 16, Lp.mb1 + half * 256,
             S_a, 256, GF_RELU);
      __syncthreads();
      gemm16(S_a, Lp.mW2 + half * 256 * 256, 256, 16,
             half == 0 ? Lp.mb2 : nullptr, S_q, 256, half == 0 ? 0 : GF_ACC);
      __syncthreads();
    }
  }

  // ---------------- outputs ----------------
  for (int e = tid; e < NBATCH * DMODEL; e += 256) P.out_q[e] = S_q[e];

  // trajectory head: h = relu(q @ tr_W1 + tr_b1) in two 256-col halves
  gemm16(S_q, P.tr_W1 + 0,   NFF, 16, P.tr_b1 + 0,   S_a, 256, GF_RELU);
  gemm16(S_q, P.tr_W1 + 256, NFF, 16, P.tr_b1 + 256, S_b, 256, GF_RELU);
  __syncthreads();
  if (tid < NBATCH * NT * 2) {
    int b = tid / 12, j = tid % 12;
    float s = P.tr_b2[j];
    for (int f = 0; f < 256; ++f) s += S_a[b * 256 + f] * P.tr_W2[f * 12 + j];
    for (int f = 0; f < 256; ++f) s += S_b[b * 256 + f] * P.tr_W2[(256 + f) * 12 + j];
    TRAJb[tid] = s;
  }
  __syncthreads();
  if (tid < NBATCH * 2) {  // cumsum over T per (batch, dim)
    int b = tid >> 1, d = tid & 1;
    float run = 0.f;
    for (int t = 0; t < NT; ++t) {
      run += TRAJb[b * 12 + t * 2 + d];
      P.out_traj[b * 12 + t * 2 + d] = run;
    }
  }
}

// ---------------------------------------------------------------------------
// Input-pointer map. Top level follows setup_inputs() insertion order:
//   0 commands, 1 bev_features, 2 track_queries, 3 motion_queries, 4.. params.
// 'params' is assumed flattened as a JAX pytree (dict keys in ASCII-sorted
// order, 'layers' list in order). Sorted top-level param keys:
//   cmd_emb, layers(3 x 24 leaves), plan_pos, pq_W1, pq_W2, pq_b1, pq_b2,
//   tr_W1, tr_W2, tr_b1, tr_b2.
// Sorted per-layer keys:
//   Wk Wo Wq Wv be1 be2 bk bo bq bv dWa dWo dWu dWv dba dbo dbu dbv g1 g2
//   mW1 mW2 mb1 mb2
// ---------------------------------------------------------------------------
extern "C" void kernel_launch(void* const* d_in, const int* in_sizes, int n_in,
                              void* d_out, int out_size, void* d_ws, size_t ws_size,
                              hipStream_t stream) {
  (void)in_sizes; (void)n_in; (void)out_size; (void)d_ws; (void)ws_size;
  auto F = [&](int i) { return (const float*)d_in[i]; };

  Params P;
  P.cmd    = (const int*)d_in[0];
  P.bev    = F(1);
  P.track  = F(2);
  P.motion = F(3);
  P.cmd_emb = F(4);
  for (int l = 0; l < NLAYER; ++l) {
    int b = 5 + l * 24;
    LayerP& L = P.lyr[l];
    // Wk=b+0, Wq=b+2, bk=b+6, bq=b+8 are dead (softmax over one key == 1).
    L.Wo  = F(b + 1);  L.Wv  = F(b + 3);
    L.be1 = F(b + 4);  L.be2 = F(b + 5);
    L.bo  = F(b + 7);  L.bv  = F(b + 9);
    L.dWa = F(b + 10); L.dWo = F(b + 11); L.dWu = F(b + 12); L.dWv = F(b + 13);
    L.dba = F(b + 14); L.dbo = F(b + 15); L.dbu = F(b + 16); L.dbv = F(b + 17);
    L.g1  = F(b + 18); L.g2  = F(b + 19);
    L.mW1 = F(b + 20); L.mW2 = F(b + 21); L.mb1 = F(b + 22); L.mb2 = F(b + 23);
  }
  P.plan_pos = F(77);
  P.pq_W1 = F(78); P.pq_W2 = F(79); P.pq_b1 = F(80); P.pq_b2 = F(81);
  P.tr_W1 = F(82); P.tr_W2 = F(83); P.tr_b1 = F(84); P.tr_b2 = F(85);
  P.out_q    = (float*)d_out;
  P.out_traj = (float*)d_out + NBATCH * DMODEL;

  // Strict layer-to-layer dependency with M=16 total rows: a single
  // 8-wave workgroup carrying all batches through the WMMA pipeline.
  hipLaunchKernelGGL(PlanFormer_18992345383477_kernel,
                     dim3(1), dim3(256), 0, stream, P);
}